// SGNSModel_68358699483146
// MI455X (gfx1250) — compile-verified
//
#include <hip/hip_runtime.h>

typedef __attribute__((ext_vector_type(2))) float v2f;
typedef __attribute__((ext_vector_type(8))) float v8f;

#define EMB     300
#define BATCH   4096
#define C_CTX   10
#define NNEG    200          // C * NEG
#define NROWS   210          // C_CTX + NNEG
#define NTILES  14           // ceil(210 / 16)
#define CLAMP_V 1.0e10f

__device__ __forceinline__ float log_sigmoid(float x) {
    // stable: min(x,0) - log(1 + exp(-|x|)); arg of log is in (1,2] so the
    // fast v_log_f32 path is accurate here (no cancellation).
    return fminf(x, 0.0f) - __logf(1.0f + __expf(-fabsf(x)));
}

__global__ __launch_bounds__(128) void sgns_score_kernel(
    const int* __restrict__ iword,
    const int* __restrict__ owords,
    const int* __restrict__ nwords,
    const float* __restrict__ W_in,
    const float* __restrict__ W_out,
    float* __restrict__ per_sample)
{
    __shared__ float lds_iv[EMB];
    __shared__ float lds_part[8];    // 4 waves x {low-half, high-half}

    const int b    = blockIdx.x;
    const int tid  = threadIdx.x;
    const int lane = tid & 31;
    const int wave = tid >> 5;

    // Stage iv = W_in[iword[b]] into LDS (1.2 KB).
    const float* ivp = W_in + (size_t)iword[b] * EMB;
    for (int i = tid; i < EMB; i += 128) lds_iv[i] = ivp[i];
    if (tid < 8) lds_part[tid] = 0.0f;
    __syncthreads();

    const int half = lane >> 4;   // 0: lanes 0-15, 1: lanes 16-31
    const int koff = half * 2;    // K sub-offset per ISA A/B f32 layout

    float acc = 0.0f;             // meaningful on lanes 0 and 16

    for (int tile = wave; tile < NTILES; tile += 4) {
        // Each lane gathers one row of the 16-row A tile.
        int row = tile * 16 + (lane & 15);
        int idx;
        if (row < C_CTX)      idx = owords[b * C_CTX + row];
        else if (row < NROWS) idx = nwords[b * NNEG + (row - C_CTX)];
        else                  idx = 0;              // pad: W_out[0] is a zero row
        const float* rowp = W_out + (size_t)idx * EMB;

        v8f c = {};
        #pragma unroll 5
        for (int k = 0; k < EMB; k += 4) {
            // A 16x4 f32: lane<16 -> K=k,k+1 ; lane>=16 -> K=k+2,k+3 (2 VGPRs)
            v2f a  = *(const v2f*)(rowp + k + koff);
            // B 4x16 f32: iv chunk broadcast to all 16 columns (same layout in K)
            v2f bb = *(const v2f*)(&lds_iv[k + koff]);
            c = __builtin_amdgcn_wmma_f32_16x16x4_f32(
                    false, a, false, bb, (short)0, c, false, false);
        }

        // D layout: lane n<16 holds rows tile*16+0..7 in c[0..7] (all columns
        // identical since B's columns are identical); lanes>=16 hold rows
        // tile*16+8..15.
        int base = tile * 16 + half * 8;
        float local = 0.0f;
        #pragma unroll
        for (int i = 0; i < 8; ++i) {
            int r = base + i;
            if (r < NROWS) {
                float s = c[i];
                float x = (r < C_CTX) ? s : -s;   // negatives: nv = -W_out row
                local += log_sigmoid(x);
            }
        }
        acc += local;
    }

    if (lane == 0)       lds_part[wave * 2 + 0] = acc;
    else if (lane == 16) lds_part[wave * 2 + 1] = acc;
    __syncthreads();

    if (tid == 0) {
        float total = 0.0f;
        #pragma unroll
        for (int i = 0; i < 8; ++i) total += lds_part[i];
        float loss = total * (1.0f / (float)C_CTX);          // oloss + nloss
        loss = fminf(fmaxf(loss, -CLAMP_V), CLAMP_V);        // clip
        per_sample[b] = loss;
    }
}

__global__ __launch_bounds__(256) void sgns_reduce_kernel(
    const float* __restrict__ per_sample, float* __restrict__ out)
{
    __shared__ float sm[256];
    const int tid = threadIdx.x;
    float s = 0.0f;
    for (int i = tid; i < BATCH; i += 256) s += per_sample[i];  // fixed order
    sm[tid] = s;
    __syncthreads();
    for (int w = 128; w > 0; w >>= 1) {
        if (tid < w) sm[tid] += sm[tid + w];
        __syncthreads();
    }
    if (tid == 0) out[0] = -sm[0] * (1.0f / (float)BATCH);
}

extern "C" void kernel_launch(void* const* d_in, const int* in_sizes, int n_in,
                              void* d_out, int out_size, void* d_ws, size_t ws_size,
                              hipStream_t stream) {
    const int*   iword  = (const int*)d_in[0];
    const int*   owords = (const int*)d_in[1];
    const int*   nwords = (const int*)d_in[2];
    const float* W_in   = (const float*)d_in[3];
    const float* W_out  = (const float*)d_in[4];
    float* per_sample = (float*)d_ws;   // BATCH floats of scratch

    sgns_score_kernel<<<BATCH, 128, 0, stream>>>(
        iword, owords, nwords, W_in, W_out, per_sample);
    sgns_reduce_kernel<<<1, 256, 0, stream>>>(per_sample, (float*)d_out);
}